// RelativeToAbsoluteModel_20882130993458
// MI455X (gfx1250) — compile-verified
//
#include <hip/hip_runtime.h>

typedef float v2f __attribute__((ext_vector_type(2)));
typedef float v8f __attribute__((ext_vector_type(8)));

#define EPS_F 0.05f
#define SEG   2048   // elements per block (8 waves x 256-element WMMA tile)

__device__ __forceinline__ float readlane_f(float v, int l) {
    return __int_as_float(__builtin_amdgcn_readlane(__float_as_int(v), l));
}

// change = table[cat], cat in 1..5 (cat 0 never occurs)
__device__ __forceinline__ float lut5(int cat, float c1, float c2, float c3, float c4, float c5) {
    float r = c5;
    r = (cat == 4) ? c4 : r;
    r = (cat == 3) ? c3 : r;
    r = (cat == 2) ? c2 : r;
    r = (cat == 1) ? c1 : r;
    return r;
}

__device__ __forceinline__ void make_table(const float* bd_p, const float* d_p, const float* s_p,
                                           const float* inc_p, const float* bi_p,
                                           float& c1, float& c2, float& c3, float& c4, float& c5) {
    float bd = *bd_p, d = *d_p, s = *s_p, inc = *inc_p, bi = *bi_p;
    c1 = fminf(bd, fminf(0.0f, d)) - EPS_F;              // big decrease
    c2 = fminf(fmaxf(d, bd + EPS_F), 0.0f - EPS_F);      // decrease (clip)
    c3 = s;                                              // same
    c4 = fminf(fmaxf(inc, 0.0f + EPS_F), bi - EPS_F);    // increase (clip)
    c5 = fmaxf(bi, fmaxf(0.0f, inc) + EPS_F);            // big increase
}

// ---------------- Kernel 1: per-block (2048-elem) sums -> partials ----------------
__global__ void k1_block_sums(const int* __restrict__ ann,
                              const float* bd_p, const float* d_p, const float* s_p,
                              const float* inc_p, const float* bi_p,
                              float* __restrict__ partials) {
    __shared__ float red[256];
    float c1, c2, c3, c4, c5;
    make_table(bd_p, d_p, s_p, inc_p, bi_p, c1, c2, c3, c4, c5);

    const int tid = threadIdx.x;
    const long base = (long)blockIdx.x * SEG + (long)tid * 8;
    const int4* p = (const int4*)(ann + base);
    int4 a = p[0];
    int4 b = p[1];
    float s = lut5(a.x, c1, c2, c3, c4, c5) + lut5(a.y, c1, c2, c3, c4, c5)
            + lut5(a.z, c1, c2, c3, c4, c5) + lut5(a.w, c1, c2, c3, c4, c5)
            + lut5(b.x, c1, c2, c3, c4, c5) + lut5(b.y, c1, c2, c3, c4, c5)
            + lut5(b.z, c1, c2, c3, c4, c5) + lut5(b.w, c1, c2, c3, c4, c5);
    red[tid] = s;
    __syncthreads();
    for (int st = 128; st > 0; st >>= 1) {
        if (tid < st) red[tid] += red[tid + st];
        __syncthreads();
    }
    if (tid == 0) partials[blockIdx.x] = red[0];
}

// ---------------- Kernel 2: exclusive scan of nb partials (single block) ----------------
__global__ void k2_scan_partials(const float* __restrict__ partials,
                                 float* __restrict__ scanned, int nb) {
    __shared__ float sums[1024];
    const int t = threadIdx.x;
    const int per = nb / 1024;   // 16 for N = 2^25
    float vals[16];
    float acc = 0.0f;
    for (int i = 0; i < per; i++) {
        vals[i] = partials[t * per + i];
        acc += vals[i];
    }
    sums[t] = acc;
    __syncthreads();
    float off = 0.0f;
    for (int j = 0; j < t; j++) off += sums[j];
    float run = off;
    for (int i = 0; i < per; i++) {
        scanned[t * per + i] = run;     // exclusive prefix of block partials
        run += vals[i];
    }
}

// ---------------- Kernel 3: tile prefix via V_WMMA_F32_16X16X4_F32 + fixups ----------------
// Each wave32 owns a 256-element tile viewed as a row-major 16x16 matrix X.
// Row-inclusive prefix = X * U (U upper-triangular ones), done as 4 chained
// v_wmma_f32_16x16x4_f32 with A = X[:,4k:4k+4] and B = U[4k:4k+4,:] built in-register.
__global__ void k3_scan_write(const int* __restrict__ ann,
                              const float* orig_p,
                              const float* bd_p, const float* d_p, const float* s_p,
                              const float* inc_p, const float* bi_p,
                              const float* __restrict__ scanned,
                              float* __restrict__ out) {
    __shared__ float waveTot[8];

    const int tid   = threadIdx.x;
    const int lane  = tid & 31;
    const int wave  = tid >> 5;
    const int n     = lane & 15;   // column (N) for B/C/D operands
    const int khalf = lane >> 4;   // upper lane group handles K+2 / M+8 halves

    float c1, c2, c3, c4, c5;
    make_table(bd_p, d_p, s_p, inc_p, bi_p, c1, c2, c3, c4, c5);

    const long blockBase = (long)blockIdx.x * SEG;
    const long tileBase  = blockBase + (long)wave * 256;

    v8f acc = {0.f, 0.f, 0.f, 0.f, 0.f, 0.f, 0.f, 0.f};

#pragma unroll
    for (int kk = 0; kk < 4; kk++) {
        // A-matrix 16x4 layout: lane(0-15) holds X[M=lane][K=4kk+{0,1}],
        //                       lane(16-31) holds X[M=lane-16][K=4kk+{2,3}]
        const int  kx  = kk * 4 + khalf * 2;
        const long idx = tileBase + (long)n * 16 + kx;
        int ca = ann[idx];
        int cb = ann[idx + 1];
        v2f A;
        A.x = lut5(ca, c1, c2, c3, c4, c5);
        A.y = lut5(cb, c1, c2, c3, c4, c5);
        // B-matrix 4x16 rows of U: VGPR0 = rows {0,2}, VGPR1 = rows {1,3} (low/high lanes)
        v2f B;
        B.x = (kx     <= n) ? 1.0f : 0.0f;
        B.y = (kx + 1 <= n) ? 1.0f : 0.0f;
        acc = __builtin_amdgcn_wmma_f32_16x16x4_f32(
            /*neg_a=*/false, A, /*neg_b=*/false, B,
            /*c_mod=*/(short)0, acc, /*reuse_a=*/false, /*reuse_b=*/false);
    }

    // Row totals T[r] = P[r][15]; D layout: vgpr v, lanes0-15 -> M=v, lanes16-31 -> M=v+8
    float t[16];
#pragma unroll
    for (int r = 0; r < 8; r++) t[r]     = readlane_f(acc[r], 15);
#pragma unroll
    for (int r = 0; r < 8; r++) t[8 + r] = readlane_f(acc[r], 31);

    float off[16];
    off[0] = 0.0f;
#pragma unroll
    for (int r = 1; r < 16; r++) off[r] = off[r - 1] + t[r - 1];
    const float tileTotal = off[15] + t[15];

    if (lane == 0) waveTot[wave] = tileTotal;
    __syncthreads();
    float wbase = 0.0f;
    for (int w = 0; w < wave; w++) wbase += waveTot[w];

    const float base = (*orig_p) + scanned[blockIdx.x] + wbase;

#pragma unroll
    for (int v = 0; v < 8; v++) {
        const int   M   = v + 8 * khalf;
        const float val = acc[v] + ((khalf == 0) ? off[v] : off[v + 8]) + base;
        out[1 + tileBase + (long)M * 16 + n] = val;
    }

    if (tid == 0 && blockIdx.x == 0) out[0] = *orig_p;
}

extern "C" void kernel_launch(void* const* d_in, const int* in_sizes, int n_in,
                              void* d_out, int out_size, void* d_ws, size_t ws_size,
                              hipStream_t stream) {
    const int*   ann  = (const int*)d_in[0];
    const float* orig = (const float*)d_in[1];
    const float* bd   = (const float*)d_in[2];
    const float* dd   = (const float*)d_in[3];
    const float* ss   = (const float*)d_in[4];
    const float* inc  = (const float*)d_in[5];
    const float* bi   = (const float*)d_in[6];
    float* out = (float*)d_out;

    const int N  = in_sizes[0];      // 2^25
    const int nb = N / SEG;          // 16384 blocks

    float* partials = (float*)d_ws;      // nb floats
    float* scanned  = partials + nb;     // nb floats (128 KB total, fits d_ws)

    k1_block_sums<<<nb, 256, 0, stream>>>(ann, bd, dd, ss, inc, bi, partials);
    k2_scan_partials<<<1, 1024, 0, stream>>>(partials, scanned, nb);
    k3_scan_write<<<nb, 256, 0, stream>>>(ann, orig, bd, dd, ss, inc, bi, scanned, out);
}